// Conv_LI_69277822485151
// MI455X (gfx1250) — compile-verified
//
#include <hip/hip_runtime.h>
#include <stdint.h>

// Fused LI-recurrence + 5x5 conv for [T=256, 1, 512, 512] fp32 on gfx1250.
//
// - Memory-bound overall (~6.7 FLOP/B): single fused pass, x streamed into LDS
//   with CDNA5 async global->LDS loads (ASYNCcnt), 4-deep pipeline.
// - Per-pixel (v,i) LI state lives in registers across all 256 timesteps.
// - The 5x5 'same' conv is executed on the XDL pipe as a banded matmul with
//   full-precision V_WMMA_F32_16X16X4_F32:
//       Out(16x16) = sum_dy  Vext(16x20) x B_dy(20x16),
//   B_dy[u][w] = k[dy][u-w] (banded), K=20 split into five K=4 WMMA chunks.
//   This offloads the 3.36 GFLOP stencil from VALU (where it would be
//   co-limiting with the 23.3 TB/s HBM stream) and co-executes with the
//   recurrence VALU work.

#define T_STEPS 256
#define IMG_H   512
#define IMG_W   512
#define BW      64            // output tile width per block
#define BH      16            // output tile height per block
#define EW      (BW + 4)      // halo-extended tile width  (68)
#define EH      (BH + 4)      // halo-extended tile height (20)
#define E_TOT   (EW * EH)     // 1360 extended pixels
#define EPAD    1536          // padded so every lane always has a legal LDS slot
#define NTHREADS 512
#define NSLOTS  3             // EPAD / NTHREADS
#define DEPTH   4             // async pipeline depth (prefetch 3 timesteps ahead)

#if defined(__HIP_DEVICE_COMPILE__) && defined(__gfx1250__) && \
    __has_builtin(__builtin_amdgcn_wmma_f32_16x16x4_f32)
#define USE_WMMA 1
#else
#define USE_WMMA 0
#endif

typedef __attribute__((ext_vector_type(2))) float v2f;
typedef __attribute__((ext_vector_type(8))) float v8f;

// CDNA5 async copy: global -> LDS, tracked by ASYNCcnt (no VGPR data path).
__device__ __forceinline__ void async_copy_b32(uint32_t lds_addr,
                                               const float* sbase,
                                               uint32_t gbyte_off) {
  asm volatile("global_load_async_to_lds_b32 %0, %1, %2"
               :: "v"(lds_addr), "v"(gbyte_off), "s"(sbase)
               : "memory");
}

// Wait until at most `ahead` batches (3 async-load instructions each) remain
// outstanding for this wave. `ahead` is workgroup-uniform.
__device__ __forceinline__ void wait_async_batches(int ahead) {
  if (ahead >= 3)      asm volatile("s_wait_asynccnt 9" ::: "memory");
  else if (ahead == 2) asm volatile("s_wait_asynccnt 6" ::: "memory");
  else if (ahead == 1) asm volatile("s_wait_asynccnt 3" ::: "memory");
  else                 asm volatile("s_wait_asynccnt 0" ::: "memory");
}

__global__ __launch_bounds__(NTHREADS)
void li_conv_fused(const float* __restrict__ x,
                   const float* __restrict__ kern,
                   float* __restrict__ out) {
  __shared__ float xb[DEPTH][EPAD];               // streamed x, quad-buffered (24 KB)
  __shared__ __align__(16) float vb[E_TOT + 48];  // membrane tile incl. halo (5.6 KB)

  const int tid  = threadIdx.x;
  const int lane = tid & 31;
  const int wid  = tid >> 5;                      // wave id 0..15 (wave32)
  const int bx   = blockIdx.x, by = blockIdx.y;
  const int x0   = bx * BW - 2;                   // extended-tile origin in the image
  const int y0   = by * BH - 2;

  // Per-slot precompute: clamped global byte offset within one frame, and
  // whether this extended pixel really lies inside the image (zero padding).
  uint32_t pixb[NSLOTS];
  bool     inb[NSLOTS];
#pragma unroll
  for (int s = 0; s < NSLOTS; ++s) {
    int e  = tid + s * NTHREADS;
    int ec = (e < E_TOT) ? e : (E_TOT - 1);       // safe address for pad lanes
    int ex = ec % EW, ey = ec / EW;
    int gx = x0 + ex, gy = y0 + ey;
    inb[s] = (e < E_TOT) && (gx >= 0) && (gx < IMG_W) && (gy >= 0) && (gy < IMG_H);
    int gxc = gx < 0 ? 0 : (gx > IMG_W - 1 ? IMG_W - 1 : gx);
    int gyc = gy < 0 ? 0 : (gy > IMG_H - 1 ? IMG_H - 1 : gy);
    pixb[s] = (uint32_t)(gyc * IMG_W + gxc) * 4u;
  }

  // LDS allocation-relative byte address of xb (generic pointer low 32 bits).
  const uint32_t xb_lds0 = (uint32_t)(size_t)&xb[0][0];

  // 5x5 weights: uniform address -> scalar loads, held for the whole kernel.
  float kw[25];
#pragma unroll
  for (int j = 0; j < 25; ++j) kw[j] = kern[j];

#if USE_WMMA
  // t-invariant banded B operands, built once into registers.
  // f32 B(4x16) lane layout (mirrors the ISA f32 A layout): lanes 0-15 carry
  // N=0..15 with {VGPR0,VGPR1}={K0,K1}; lanes 16-31 carry {K2,K3}.
  // Chunk (dy,c): B[u][n] = k[dy][u-n] for 0<=u-n<=4 else 0, u = 4c + 2*kh + j.
  const int bn  = lane & 15;
  const int bkh = lane >> 4;
  float Bop[5][5][2];
#pragma unroll
  for (int dy = 0; dy < 5; ++dy)
#pragma unroll
    for (int c = 0; c < 5; ++c)
#pragma unroll
      for (int j = 0; j < 2; ++j) {
        int g = 4 * c + 2 * bkh + j - bn;         // u - n
        float val = 0.0f;
#pragma unroll
        for (int g5 = 0; g5 < 5; ++g5) val = (g == g5) ? kw[dy * 5 + g5] : val;
        Bop[dy][c][j] = val;
      }
#endif

  float v[NSLOTS], cur[NSLOTS];
#pragma unroll
  for (int s = 0; s < NSLOTS; ++s) { v[s] = 0.0f; cur[s] = 0.0f; }

  // Prologue: fill the async pipeline with timesteps 0 .. DEPTH-2.
#pragma unroll
  for (int p = 0; p < DEPTH - 1; ++p) {
    uint32_t go = (uint32_t)p * (uint32_t)(IMG_H * IMG_W * 4);
    uint32_t lb = xb_lds0 + (uint32_t)(p * EPAD * 4);
#pragma unroll
    for (int s = 0; s < NSLOTS; ++s)
      async_copy_b32(lb + (uint32_t)(tid + s * NTHREADS) * 4u, x, go + pixb[s]);
  }

  for (int t = 0; t < T_STEPS; ++t) {
    // Keep the pipeline full: issue the batch for t+DEPTH-1 before waiting,
    // so HBM latency overlaps this step's compute.
    const int tp = t + (DEPTH - 1);
    if (tp < T_STEPS) {
      uint32_t go = (uint32_t)tp * (uint32_t)(IMG_H * IMG_W * 4);
      uint32_t lb = xb_lds0 + (uint32_t)((tp & (DEPTH - 1)) * EPAD * 4);
#pragma unroll
      for (int s = 0; s < NSLOTS; ++s)
        async_copy_b32(lb + (uint32_t)(tid + s * NTHREADS) * 4u, x, go + pixb[s]);
    }
    int ahead = T_STEPS - 1 - t;
    if (ahead > DEPTH - 1) ahead = DEPTH - 1;
    wait_async_batches(ahead);      // my batch for step t has landed in LDS
    __syncthreads();                // everyone's batch t visible; vb free to overwrite

    // --- LI state update (dt*tau_mem_inv == 1.0, dt*tau_syn_inv == 0.2),
    //     fp op order mirrors the reference ---
    const float* xt = &xb[t & (DEPTH - 1)][0];
#pragma unroll
    for (int s = 0; s < NSLOTS; ++s) {
      int e = tid + s * NTHREADS;
      if (e < E_TOT) {
        float xv = xt[e];
        float vn = v[s] + ((0.0f - v[s]) + cur[s]);   // v' = v + 1.0*((v_leak-v)+i)
        cur[s]   = (cur[s] - 0.2f * cur[s]) + xv;     // i' = i - 0.2*i + x_t
        v[s]     = vn;
        vb[e]    = inb[s] ? vn : 0.0f;                // zeros implement 'same' padding
      }
    }
    __syncthreads();                // vb complete for this timestep

#if USE_WMMA
    // --- conv on the XDL pipe: waves 0..3 each own one 16x16 subtile ---
    if (wid < 4) {                  // wave-uniform branch -> EXEC all-ones inside
      const int m     = lane & 15;  // A-matrix row (M)
      const int kh    = lane >> 4;  // K half-select (K0/K1 vs K2/K3)
      const int ubase = wid * 16 + kh * 2;
      v8f acc = {};
#pragma unroll
      for (int dy = 0; dy < 5; ++dy) {
        const float* arow = &vb[(dy + m) * EW + ubase];
#pragma unroll
        for (int c = 0; c < 5; ++c) {
          v2f A = *(const v2f*)(arow + 4 * c);     // ds_load_b64, 8B aligned
          v2f B; B.x = Bop[dy][c][0]; B.y = Bop[dy][c][1];
          acc = __builtin_amdgcn_wmma_f32_16x16x4_f32(
              false, A, false, B, (short)0, acc, false, false);
        }
      }
      // D layout: VGPR r -> rows M=r (lanes 0-15) and M=r+8 (lanes 16-31).
      float* op = out + (size_t)t * (size_t)(IMG_H * IMG_W)
                      + (size_t)(by * BH + (kh ? 8 : 0)) * IMG_W
                      + (size_t)(bx * BW + wid * 16 + m);
#pragma unroll
      for (int r = 0; r < 8; ++r)
        __builtin_nontemporal_store(acc[r], op + (size_t)r * IMG_W);
    }
#else
    // --- VALU fallback: 5x5 stencil from LDS, 2 output pixels per thread ---
    const size_t obase = (size_t)t * (size_t)(IMG_H * IMG_W);
#pragma unroll
    for (int r = 0; r < 2; ++r) {
      int p  = tid + r * NTHREADS;
      int ox = p & (BW - 1);
      int oy = p >> 6;
      float acc = 0.0f;
#pragma unroll
      for (int dy = 0; dy < 5; ++dy)
#pragma unroll
        for (int dx = 0; dx < 5; ++dx)
          acc = fmaf(kw[5 * dy + dx], vb[(oy + dy) * EW + (ox + dx)], acc);
      __builtin_nontemporal_store(
          acc, &out[obase + (size_t)(by * BH + oy) * IMG_W + (size_t)(bx * BW + ox)]);
    }
#endif
  }
}

extern "C" void kernel_launch(void* const* d_in, const int* in_sizes, int n_in,
                              void* d_out, int out_size, void* d_ws, size_t ws_size,
                              hipStream_t stream) {
  (void)in_sizes; (void)n_in; (void)out_size; (void)d_ws; (void)ws_size;
  const float* x    = (const float*)d_in[0];   // [256,1,512,512] f32
  const float* kern = (const float*)d_in[1];   // [5,5] f32
  float*       out  = (float*)d_out;           // [256,1,512,512] f32

  dim3 grid(IMG_W / BW, IMG_H / BH, 1);        // 8 x 32 = 256 blocks
  dim3 block(NTHREADS, 1, 1);                  // 16 wave32s
  li_conv_fused<<<grid, block, 0, stream>>>(x, kern, out);
}